// TransformerBlock_6107443495405
// MI455X (gfx1250) — compile-verified
//
#include <hip/hip_runtime.h>
#include <hip/hip_bf16.h>
#include <math.h>

// ---------------------------------------------------------------------------
// Restormer block for MI455X (gfx1250, wave32).
// Memory-bound (~1.5GB traffic @ 23.3TB/s ~ 65us; only ~61 GFLOP), so:
//  - fp32 end-to-end, V_WMMA_F32_16X16X4_F32 for the big 1x1-conv GEMMs
//  - residual adds fused into GEMM epilogues
//  - GDFN dwconv + GELU-gate fused into one pass
//  - attention Gram computed on raw q/k; 1/||.|| folded into softmax scaling
// GEMM is templated on compile-time shapes so the k-loop is branch-free:
// unconditional global_load_b64/b32 + 4x v_wmma per k-step (32Mx32N per wave).
// ---------------------------------------------------------------------------

#define B_     4
#define C_     192
#define H_     128
#define W_     128
#define HW_    16384
#define HEADS_ 8
#define HD_    24          // C_/HEADS_
#define C3_    576         // 3*C_
#define HID_   510
#define HID2_  1020
#define LN_EPS 1e-5f
#define NRM_EPS 1e-12f

typedef float v2f __attribute__((ext_vector_type(2)));
typedef float v8f __attribute__((ext_vector_type(8)));

// ---------------------------------------------------------------------------
// LayerNorm over channel dim, NCHW. One thread per pixel; channel loop is
// strided by HW so lanes are coalesced along hw.
// ---------------------------------------------------------------------------
__global__ void ln_kernel(const float* __restrict__ x,
                          const float* __restrict__ g,
                          const float* __restrict__ b,
                          float* __restrict__ y) {
  int batch = blockIdx.y;
  int p = blockIdx.x * blockDim.x + threadIdx.x;       // pixel in [0,HW)
  const float* xb = x + (size_t)batch * C_ * HW_ + p;
  float s = 0.f, s2 = 0.f;
  for (int c = 0; c < C_; ++c) {
    float v = xb[(size_t)c * HW_];
    s += v; s2 += v * v;
  }
  float mean = s * (1.f / C_);
  float var  = s2 * (1.f / C_) - mean * mean;
  float inv  = rsqrtf(var + LN_EPS);
  float* yb = y + (size_t)batch * C_ * HW_ + p;
  for (int c = 0; c < C_; ++c) {
    yb[(size_t)c * HW_] = (xb[(size_t)c * HW_] - mean) * inv * g[c] + b[c];
  }
}

// ---------------------------------------------------------------------------
// 1x1-conv GEMM with fp32 WMMA:  Y[z][m][n] = sum_k W[m][k]*X[z][k][n]
// (+ R[z][m][n] if RES).
// Compile-time M,K,N. Wave tile: 32(M) x 32(N) = 4 wmma accumulators.
// Block = 8 waves as 2(Mw) x 4(Nw) -> 64(M) x 128(N) block tile.
// A (16x4 f32) layout: lane(0..15)=M row, vgpr0/1 = K{0,1}; lanes16..31 K{2,3}
//   -> per-lane contiguous float2 at W[m][k + 2*laneHi].
// B (4x16 f32): vgpr0/1 = K rows {0,1} (+2 for lanes 16..31), lane&15 = N col.
// C/D: vgpr v -> m = v + 8*(lane>>4), n = lane&15.
// MG: clamp A rows (loads always in-bounds), guard stores only.
// ---------------------------------------------------------------------------
template <int M, int K, int N, bool MG, bool RES>
__global__ void conv1x1_wmma(const float* __restrict__ X,
                             const float* __restrict__ Wt,
                             const float* __restrict__ R,
                             float* __restrict__ Y) {
  const int lane   = threadIdx.x & 31;
  const int wave   = threadIdx.x >> 5;
  const int laneLo = lane & 15;
  const int laneHi = lane >> 4;
  const int mBlock = blockIdx.y * 64 + (wave >> 2) * 32;   // M tiles: mBlock, +16
  const int nBlock = blockIdx.x * 128 + (wave & 3) * 32;   // N tiles: nBlock, +16

  int mA0 = mBlock + laneLo;
  int mA1 = mBlock + 16 + laneLo;
  if (MG) { mA0 = min(mA0, M - 1); mA1 = min(mA1, M - 1); }

  const float* w0  = Wt + (size_t)mA0 * K + 2 * laneHi;
  const float* w1  = Wt + (size_t)mA1 * K + 2 * laneHi;
  const float* xb0 = X + (size_t)blockIdx.z * K * N +
                     (size_t)(2 * laneHi) * N + nBlock + laneLo;

  v8f acc00 = {}, acc01 = {}, acc10 = {}, acc11 = {};
  constexpr int KMAIN = K & ~3;

#pragma unroll 2
  for (int k = 0; k < KMAIN; k += 4) {
    v2f a0 = *(const v2f*)(w0 + k);
    v2f a1 = *(const v2f*)(w1 + k);
    const float* xk = xb0 + (size_t)k * N;
    v2f b0, b1;
    b0.x = xk[0];
    b0.y = xk[N];
    b1.x = xk[16];
    b1.y = xk[N + 16];
    acc00 = __builtin_amdgcn_wmma_f32_16x16x4_f32(false, a0, false, b0,
                                                  (short)0, acc00, false, false);
    acc01 = __builtin_amdgcn_wmma_f32_16x16x4_f32(false, a0, false, b1,
                                                  (short)0, acc01, false, false);
    acc10 = __builtin_amdgcn_wmma_f32_16x16x4_f32(false, a1, false, b0,
                                                  (short)0, acc10, false, false);
    acc11 = __builtin_amdgcn_wmma_f32_16x16x4_f32(false, a1, false, b1,
                                                  (short)0, acc11, false, false);
  }

  if constexpr ((K & 3) != 0) {
    // K tail (K=510): rows KMAIN,KMAIN+1 valid; laneHi==1 rows are >= K -> 0
    const bool hi = (laneHi != 0);
    const float* xk = xb0 + (size_t)KMAIN * N;
    v2f a0, a1, b0, b1;
    a0.x = hi ? 0.f : w0[KMAIN];
    a0.y = hi ? 0.f : w0[KMAIN + 1];
    a1.x = hi ? 0.f : w1[KMAIN];
    a1.y = hi ? 0.f : w1[KMAIN + 1];
    b0.x = hi ? 0.f : xk[0];
    b0.y = hi ? 0.f : xk[N];
    b1.x = hi ? 0.f : xk[16];
    b1.y = hi ? 0.f : xk[N + 16];
    acc00 = __builtin_amdgcn_wmma_f32_16x16x4_f32(false, a0, false, b0,
                                                  (short)0, acc00, false, false);
    acc01 = __builtin_amdgcn_wmma_f32_16x16x4_f32(false, a0, false, b1,
                                                  (short)0, acc01, false, false);
    acc10 = __builtin_amdgcn_wmma_f32_16x16x4_f32(false, a1, false, b0,
                                                  (short)0, acc10, false, false);
    acc11 = __builtin_amdgcn_wmma_f32_16x16x4_f32(false, a1, false, b1,
                                                  (short)0, acc11, false, false);
  }

  const size_t yoff = (size_t)blockIdx.z * M * N;
#pragma unroll
  for (int t = 0; t < 4; ++t) {
    const int mT = mBlock + (t >> 1) * 16;
    const int nT = nBlock + (t & 1) * 16 + laneLo;
    const v8f& a = (t == 0) ? acc00 : (t == 1) ? acc01 : (t == 2) ? acc10 : acc11;
#pragma unroll
    for (int v = 0; v < 8; ++v) {
      int m = mT + v + 8 * laneHi;
      if (!MG || m < M) {
        size_t idx = yoff + (size_t)m * N + nT;
        float val = a[v];
        if constexpr (RES) val += R[idx];
        Y[idx] = val;
      }
    }
  }
}

// ---------------------------------------------------------------------------
// Depthwise 3x3, pad=1, LDS-tiled (16x16 output tile, 18x18 halo).
// grid.z = batch*CH + ch
// ---------------------------------------------------------------------------
__global__ void dwconv3x3_kernel(const float* __restrict__ X,
                                 const float* __restrict__ Wd,
                                 float* __restrict__ Y, int CH) {
  __shared__ float tile[18][20];
  const int ch = blockIdx.z % CH;
  const size_t base = (size_t)blockIdx.z * HW_;
  const int x0 = blockIdx.x * 16, y0 = blockIdx.y * 16;

  for (int i = threadIdx.x; i < 18 * 18; i += 256) {
    int lx = i % 18, ly = i / 18;
    int gx = x0 + lx - 1, gy = y0 + ly - 1;
    float v = 0.f;
    if (gx >= 0 && gx < W_ && gy >= 0 && gy < H_)
      v = X[base + (size_t)gy * W_ + gx];
    tile[ly][lx] = v;
  }
  __syncthreads();

  const float* w = Wd + (size_t)ch * 9;
  int tx = threadIdx.x & 15, ty = threadIdx.x >> 4;
  float acc = 0.f;
#pragma unroll
  for (int dy = 0; dy < 3; ++dy)
#pragma unroll
    for (int dx = 0; dx < 3; ++dx)
      acc = fmaf(tile[ty + dy][tx + dx], w[dy * 3 + dx], acc);
  Y[base + (size_t)(y0 + ty) * W_ + (x0 + tx)] = acc;
}

// ---------------------------------------------------------------------------
// GDFN fused: dwconv3x3 on channel pair (ch, ch+HID) of the 2*HID hidden
// tensor, then g = gelu_exact(dw1) * dw2.  grid.z = batch*HID + ch
// ---------------------------------------------------------------------------
__device__ __forceinline__ float gelu_exact(float x) {
  return 0.5f * x * (1.f + erff(x * 0.70710678118654752f));
}

__global__ void dwconv_glu_kernel(const float* __restrict__ X,  // [B][HID2][HW]
                                  const float* __restrict__ Wd, // [HID2][9]
                                  float* __restrict__ G) {      // [B][HID][HW]
  __shared__ float t1[18][20];
  __shared__ float t2[18][20];
  const int ch = blockIdx.z % HID_;
  const int bt = blockIdx.z / HID_;
  const size_t base1 = ((size_t)bt * HID2_ + ch) * HW_;
  const size_t base2 = ((size_t)bt * HID2_ + ch + HID_) * HW_;
  const int x0 = blockIdx.x * 16, y0 = blockIdx.y * 16;

  for (int i = threadIdx.x; i < 18 * 18; i += 256) {
    int lx = i % 18, ly = i / 18;
    int gx = x0 + lx - 1, gy = y0 + ly - 1;
    float v1 = 0.f, v2 = 0.f;
    if (gx >= 0 && gx < W_ && gy >= 0 && gy < H_) {
      size_t o = (size_t)gy * W_ + gx;
      v1 = X[base1 + o];
      v2 = X[base2 + o];
    }
    t1[ly][lx] = v1;
    t2[ly][lx] = v2;
  }
  __syncthreads();

  const float* w1 = Wd + (size_t)ch * 9;
  const float* w2 = Wd + (size_t)(ch + HID_) * 9;
  int tx = threadIdx.x & 15, ty = threadIdx.x >> 4;
  float a1 = 0.f, a2 = 0.f;
#pragma unroll
  for (int dy = 0; dy < 3; ++dy)
#pragma unroll
    for (int dx = 0; dx < 3; ++dx) {
      a1 = fmaf(t1[ty + dy][tx + dx], w1[dy * 3 + dx], a1);
      a2 = fmaf(t2[ty + dy][tx + dx], w2[dy * 3 + dx], a2);
    }
  G[((size_t)bt * HID_ + ch) * HW_ + (size_t)(y0 + ty) * W_ + (x0 + tx)] =
      gelu_exact(a1) * a2;
}

// ---------------------------------------------------------------------------
// 1/max(||row||2, eps) over HW for the 2*C q,k channels of qkv_dw.
// grid.x = batch*384 + ch (ch in [0,384): q then k). Block tree-reduce in LDS.
// ---------------------------------------------------------------------------
__global__ void rnorm_kernel(const float* __restrict__ QKV,  // [B][576][HW]
                             float* __restrict__ RN) {       // [B][384]
  __shared__ float red[256];
  const int bt = blockIdx.x / 384;
  const int ch = blockIdx.x % 384;
  const float* p = QKV + ((size_t)bt * C3_ + ch) * HW_;
  float s = 0.f;
  for (int e = threadIdx.x; e < HW_; e += 256) {
    float v = p[e];
    s = fmaf(v, v, s);
  }
  red[threadIdx.x] = s;
  __syncthreads();
  for (int off = 128; off > 0; off >>= 1) {
    if (threadIdx.x < off) red[threadIdx.x] += red[threadIdx.x + off];
    __syncthreads();
  }
  if (threadIdx.x == 0)
    RN[blockIdx.x] = 1.f / fmaxf(sqrtf(red[0]), NRM_EPS);
}

// ---------------------------------------------------------------------------
// Channel-attention scores + softmax per (batch, head).
// scores[c,d] = (sum_e k[c,e]*q[d,e]) * rk[c]*rq[d]*temp[h]; softmax over d.
// Block = 24 waves (768 thr); wave c accumulates its row, q tile staged in
// LDS, lane partials reduced with shfl_xor, softmax by threads 0..23.
// ---------------------------------------------------------------------------
__global__ void gram_softmax_kernel(const float* __restrict__ QKV, // dwconv'd
                                    const float* __restrict__ RN,
                                    const float* __restrict__ temp,
                                    float* __restrict__ ATTN) {    // [B][8][24][24]
  __shared__ float qtile[24 * 256];
  __shared__ float sS[24 * 24];
  const int bh = blockIdx.x;
  const int bt = bh / HEADS_, h = bh % HEADS_;
  const int wave = threadIdx.x >> 5;   // = c row
  const int lane = threadIdx.x & 31;
  const float* qbase = QKV + ((size_t)bt * C3_ + h * HD_) * HW_;
  const float* kptr  = QKV + ((size_t)bt * C3_ + C_ + h * HD_ + wave) * HW_;

  float acc[24];
#pragma unroll
  for (int d = 0; d < 24; ++d) acc[d] = 0.f;

  for (int e0 = 0; e0 < HW_; e0 += 256) {
    // cooperative q stage: 24*256 / 768 = 8 per thread, row-coalesced
#pragma unroll
    for (int j = 0; j < 8; ++j) {
      int idx = threadIdx.x + 768 * j;
      int d = idx >> 8, e = idx & 255;
      qtile[idx] = qbase[(size_t)d * HW_ + e0 + e];
    }
    __syncthreads();
#pragma unroll
    for (int j = 0; j < 8; ++j) {
      int e = lane + 32 * j;
      float kc = kptr[e0 + e];
#pragma unroll
      for (int d = 0; d < 24; ++d)
        acc[d] = fmaf(kc, qtile[d * 256 + e], acc[d]);
    }
    __syncthreads();
  }

#pragma unroll
  for (int d = 0; d < 24; ++d) {
    float v = acc[d];
    for (int off = 16; off > 0; off >>= 1) v += __shfl_xor(v, off, 32);
    if (lane == 0) sS[wave * 24 + d] = v;
  }
  __syncthreads();

  if (threadIdx.x < 24) {
    int c = threadIdx.x;
    float rk = RN[bt * 384 + 192 + h * HD_ + c];
    float tmp = temp[h];
    float row[24];
    float mx = -3.4e38f;
#pragma unroll
    for (int d = 0; d < 24; ++d) {
      float s = sS[c * 24 + d] * rk * RN[bt * 384 + h * HD_ + d] * tmp;
      row[d] = s;
      mx = fmaxf(mx, s);
    }
    float sum = 0.f;
#pragma unroll
    for (int d = 0; d < 24; ++d) {
      row[d] = expf(row[d] - mx);
      sum += row[d];
    }
    float inv = 1.f / sum;
    float* out = ATTN + ((size_t)bh * HD_ + c) * HD_;
#pragma unroll
    for (int d = 0; d < 24; ++d) out[d] = row[d] * inv;
  }
}

// ---------------------------------------------------------------------------
// out[c,e] = sum_d attn[c,d] * v[d,e] per (batch,head). attn in LDS, v
// cached in registers (each v element read exactly once).
// grid: (HW/256, B*HEADS). Output layout [B][C][HW].
// ---------------------------------------------------------------------------
__global__ void attn_apply_kernel(const float* __restrict__ QKV,
                                  const float* __restrict__ ATTN,
                                  float* __restrict__ OUT) {
  __shared__ float sA[24 * 24];
  const int bh = blockIdx.y;
  const int bt = bh / HEADS_, h = bh % HEADS_;
  for (int i = threadIdx.x; i < 24 * 24; i += 256)
    sA[i] = ATTN[(size_t)bh * HD_ * HD_ + i];
  __syncthreads();

  const int e = blockIdx.x * 256 + threadIdx.x;
  const float* vbase = QKV + ((size_t)bt * C3_ + 2 * C_ + h * HD_) * HW_ + e;
  float vr[24];
#pragma unroll
  for (int d = 0; d < 24; ++d) vr[d] = vbase[(size_t)d * HW_];

  float* obase = OUT + ((size_t)bt * C_ + h * HD_) * HW_ + e;
#pragma unroll
  for (int c = 0; c < 24; ++c) {
    float acc = 0.f;
#pragma unroll
    for (int d = 0; d < 24; ++d) acc = fmaf(sA[c * 24 + d], vr[d], acc);
    obase[(size_t)c * HW_] = acc;
  }
}

// ---------------------------------------------------------------------------
// host launcher
// ---------------------------------------------------------------------------
extern "C" void kernel_launch(void* const* d_in, const int* in_sizes, int n_in,
                              void* d_out, int out_size, void* d_ws, size_t ws_size,
                              hipStream_t stream) {
  const float* x       = (const float*)d_in[0];
  const float* ln1_g   = (const float*)d_in[1];
  const float* ln1_b   = (const float*)d_in[2];
  const float* qkv_w   = (const float*)d_in[3];   // [576][192]
  const float* qkv_dw  = (const float*)d_in[4];   // [576][9]
  const float* proj_w  = (const float*)d_in[5];   // [192][192]
  const float* temp    = (const float*)d_in[6];   // [8]
  const float* ln2_g   = (const float*)d_in[7];
  const float* ln2_b   = (const float*)d_in[8];
  const float* pin_w   = (const float*)d_in[9];   // [1020][192]
  const float* dw_w    = (const float*)d_in[10];  // [1020][9]
  const float* pout_w  = (const float*)d_in[11];  // [192][510]
  float* out = (float*)d_out;                     // [B][192][HW]

  // workspace layout (floats)
  float* ws = (float*)d_ws;
  const size_t SZ_XN  = (size_t)B_ * C_ * HW_;       // 12.58M
  const size_t SZ_R1  = (size_t)B_ * HID2_ * HW_;    // 66.85M (>= qkv 37.75M)
  const size_t SZ_R2  = (size_t)B_ * C3_ * HW_;      // 37.75M (>= G 33.42M)
  const size_t SZ_R3  = (size_t)B_ * C_ * HW_;       // 12.58M
  float* XN   = ws;                                   // ln out (phase1 & 2)
  float* R1   = XN + SZ_XN;                           // qkv / h12
  float* R2   = R1 + SZ_R1;                           // qkv_dw / G
  float* AOUT = R2 + SZ_R2;                           // attn output
  float* RN   = AOUT + SZ_R3;                         // [B*384]
  float* ATTN = RN + (size_t)B_ * 384;                // [B*8*24*24]
  (void)ws_size; (void)in_sizes; (void)n_in; (void)out_size;

  dim3 blk256(256);

  // ---- phase 1: MDTA ----
  ln_kernel<<<dim3(HW_ / 256, B_), blk256, 0, stream>>>(x, ln1_g, ln1_b, XN);

  conv1x1_wmma<C3_, C_, HW_, false, false>
      <<<dim3(HW_ / 128, C3_ / 64, B_), blk256, 0, stream>>>(
          XN, qkv_w, nullptr, R1);

  dwconv3x3_kernel<<<dim3(W_ / 16, H_ / 16, B_ * C3_), blk256, 0, stream>>>(
      R1, qkv_dw, R2, C3_);

  rnorm_kernel<<<dim3(B_ * 384), blk256, 0, stream>>>(R2, RN);

  gram_softmax_kernel<<<dim3(B_ * HEADS_), dim3(768), 0, stream>>>(
      R2, RN, temp, ATTN);

  attn_apply_kernel<<<dim3(HW_ / 256, B_ * HEADS_), blk256, 0, stream>>>(
      R2, ATTN, AOUT);

  // x1 = x + proj(attn_out)  -> d_out
  conv1x1_wmma<C_, C_, HW_, false, true>
      <<<dim3(HW_ / 128, C_ / 64, B_), blk256, 0, stream>>>(
          AOUT, proj_w, x, out);

  // ---- phase 2: GDFN ----
  ln_kernel<<<dim3(HW_ / 256, B_), blk256, 0, stream>>>(out, ln2_g, ln2_b, XN);

  conv1x1_wmma<HID2_, C_, HW_, true, false>
      <<<dim3(HW_ / 128, (HID2_ + 63) / 64, B_), blk256, 0, stream>>>(
          XN, pin_w, nullptr, R1);

  dwconv_glu_kernel<<<dim3(W_ / 16, H_ / 16, B_ * HID_), blk256, 0, stream>>>(
      R1, dw_w, R2);

  // out = x1 + pout(G)  (reads+writes d_out elementwise-aligned: safe)
  conv1x1_wmma<C_, HID_, HW_, false, true>
      <<<dim3(HW_ / 128, C_ / 64, B_), blk256, 0, stream>>>(
          R2, pout_w, out, out);
}